// TeamGNN_61160334295458
// MI455X (gfx1250) — compile-verified
//
#include <hip/hip_runtime.h>
#include <hip/hip_bf16.h>

// ---------------------------------------------------------------------------
// 2-layer GCN for MI455X (gfx1250).
//   h   = X @ W                     (fp32 WMMA 16x16x4)
//   hs  = h * dinv[row]             (fused GEMM epilogue; also acc = hs -> self loop)
//   acc[dst] += hs[src]             (pure atomic scatter, L2-resident)
//   out = leaky(LN(acc*dinv + b))   (wave-per-node)
//   pool: segment mean via atomics
// ---------------------------------------------------------------------------

typedef float v2f __attribute__((ext_vector_type(2)));
typedef float v8f __attribute__((ext_vector_type(8)));

#define HID        256
#define LDS_STRIDE 260     // padded row stride (floats) to avoid LDS bank conflicts
#define LEAKY      0.01f
#define LN_EPS     1e-5f

__device__ __forceinline__ void atomAdd(float* p, float v) {
  unsafeAtomicAdd(p, v);   // hardware global_atomic_add_f32
}

// --------------------------- degree / norm ---------------------------------
__global__ void k_init_deg(float* __restrict__ deg, int N) {
  int i = blockIdx.x * 256 + threadIdx.x;
  if (i < N) deg[i] = 1.0f;                       // self loop
}

__global__ void k_deg_scatter(const int* __restrict__ dst, float* __restrict__ deg, int E) {
  int e = blockIdx.x * 256 + threadIdx.x;
  if (e < E) atomAdd(&deg[dst[e]], 1.0f);
}

__global__ void k_deg_to_dinv(float* __restrict__ deg, int N) {
  int i = blockIdx.x * 256 + threadIdx.x;
  if (i < N) deg[i] = rsqrtf(deg[i]);
}

// --------------------------- fp32 WMMA GEMM --------------------------------
// C[16 x 256] tile per block; 4 waves, each wave a 16x64 slab via 4 v8f accs.
// Epilogue scales rows by dinv and stores to BOTH hs and acc (self-loop init).
__global__ __launch_bounds__(128)
void gemm_rowscale(const float* __restrict__ X, const float* __restrict__ W,
                   const float* __restrict__ dinv,
                   float* __restrict__ hs, float* __restrict__ acc) {
  __shared__ float Xs[16 * LDS_STRIDE];
  __shared__ float Ws[64 * LDS_STRIDE];

  const int tid  = threadIdx.x;
  const int wave = tid >> 5;
  const int lane = tid & 31;
  const int half = lane >> 4;       // 0: lanes 0-15, 1: lanes 16-31
  const int l16  = lane & 15;
  const int row0 = blockIdx.x << 4;
  const int n0   = wave << 6;       // 64-col slab per wave

  // Stage X tile (16 x 256) into LDS, coalesced float4.
  for (int i = tid; i < 16 * 64; i += 128) {
    const int r = i >> 6, c4 = i & 63;
    const float4 v = reinterpret_cast<const float4*>(X + (size_t)(row0 + r) * HID)[c4];
    float* p = &Xs[r * LDS_STRIDE + (c4 << 2)];
    p[0] = v.x; p[1] = v.y; p[2] = v.z; p[3] = v.w;
  }

  v8f c0 = {}, c1 = {}, c2 = {}, c3 = {};

  for (int kc = 0; kc < HID; kc += 64) {
    __syncthreads();                               // protect Xs (first) / Ws reuse
    // Stage W chunk (64 x 256) into LDS.
    for (int i = tid; i < 64 * 64; i += 128) {
      const int r = i >> 6, c4 = i & 63;
      const float4 v = reinterpret_cast<const float4*>(W + (size_t)(kc + r) * HID)[c4];
      float* p = &Ws[r * LDS_STRIDE + (c4 << 2)];
      p[0] = v.x; p[1] = v.y; p[2] = v.z; p[3] = v.w;
    }
    __syncthreads();

#pragma unroll 4
    for (int k = 0; k < 64; k += 4) {
      const int kk = k + (half << 1);   // lanes 0-15: K=k,k+1 ; lanes 16-31: K=k+2,k+3
      // A fragment (16x4 f32): lane row = l16
      v2f a;
      a.x = Xs[l16 * LDS_STRIDE + kk];
      a.y = Xs[l16 * LDS_STRIDE + kk + 1];
      // B fragments (4x16 f32): VGPR0 = row kk, VGPR1 = row kk+1, N = n0 + 16s + l16
      const float* wr0 = &Ws[kk * LDS_STRIDE];
      const float* wr1 = &Ws[(kk + 1) * LDS_STRIDE];
      v2f b0, b1, b2, b3;
      b0.x = wr0[n0 + l16];      b0.y = wr1[n0 + l16];
      b1.x = wr0[n0 + l16 + 16]; b1.y = wr1[n0 + l16 + 16];
      b2.x = wr0[n0 + l16 + 32]; b2.y = wr1[n0 + l16 + 32];
      b3.x = wr0[n0 + l16 + 48]; b3.y = wr1[n0 + l16 + 48];

      c0 = __builtin_amdgcn_wmma_f32_16x16x4_f32(false, a, false, b0, (short)0, c0, false, false);
      c1 = __builtin_amdgcn_wmma_f32_16x16x4_f32(false, a, false, b1, (short)0, c1, false, false);
      c2 = __builtin_amdgcn_wmma_f32_16x16x4_f32(false, a, false, b2, (short)0, c2, false, false);
      c3 = __builtin_amdgcn_wmma_f32_16x16x4_f32(false, a, false, b3, (short)0, c3, false, false);
    }
  }

  // Epilogue: D layout -> M = v + 8*half, N = n0 + 16s + l16. Scale rows by dinv.
  const int mbase = row0 + (half << 3);
  float dv[8];
#pragma unroll
  for (int v = 0; v < 8; ++v) dv[v] = dinv[mbase + v];
#pragma unroll
  for (int v = 0; v < 8; ++v) {
    const size_t rb = (size_t)(mbase + v) * HID + n0 + l16;
    const float e0 = c0[v] * dv[v];
    const float e1 = c1[v] * dv[v];
    const float e2 = c2[v] * dv[v];
    const float e3 = c3[v] * dv[v];
    hs[rb]      = e0; acc[rb]      = e0;
    hs[rb + 16] = e1; acc[rb + 16] = e1;
    hs[rb + 32] = e2; acc[rb + 32] = e2;
    hs[rb + 48] = e3; acc[rb + 48] = e3;
  }
}

// --------------------------- edge scatter ----------------------------------
// thread = (edge, 4-channel group): float4 gather + 4 fp32 atomics (L2-resident)
__global__ __launch_bounds__(256)
void k_edge_scatter(const int* __restrict__ src, const int* __restrict__ dst,
                    const float* __restrict__ hs, float* __restrict__ acc, int E) {
  const int idx = blockIdx.x * 256 + threadIdx.x;
  const int e = idx >> 6;
  if (e >= E) return;
  const int c4 = idx & 63;
  const int s = src[e];
  const int d = dst[e];
  const float4 v = reinterpret_cast<const float4*>(hs + (size_t)s * HID)[c4];
  float* p = acc + (size_t)d * HID + (c4 << 2);
  atomAdd(p + 0, v.x);
  atomAdd(p + 1, v.y);
  atomAdd(p + 2, v.z);
  atomAdd(p + 3, v.w);
}

// -------------------- bias + LayerNorm + LeakyReLU (wave/node) -------------
__global__ __launch_bounds__(256)
void k_ln_act(const float* __restrict__ acc, const float* __restrict__ dinv,
              const float* __restrict__ bias, const float* __restrict__ gw,
              const float* __restrict__ gb, float* __restrict__ out, int N) {
  const int node = blockIdx.x * 8 + (threadIdx.x >> 5);
  if (node >= N) return;
  const int lane = threadIdx.x & 31;
  const float di = dinv[node];
  const float* ap = acc + (size_t)node * HID;

  const float4 xa = reinterpret_cast<const float4*>(ap)[lane];        // ch 4*lane
  const float4 xb = reinterpret_cast<const float4*>(ap)[lane + 32];   // ch 128+4*lane
  const float4 ba = reinterpret_cast<const float4*>(bias)[lane];
  const float4 bb = reinterpret_cast<const float4*>(bias)[lane + 32];

  float v[8];
  v[0] = xa.x * di + ba.x;  v[1] = xa.y * di + ba.y;
  v[2] = xa.z * di + ba.z;  v[3] = xa.w * di + ba.w;
  v[4] = xb.x * di + bb.x;  v[5] = xb.y * di + bb.y;
  v[6] = xb.z * di + bb.z;  v[7] = xb.w * di + bb.w;

  float s = 0.f, s2 = 0.f;
#pragma unroll
  for (int i = 0; i < 8; ++i) { s += v[i]; s2 += v[i] * v[i]; }
#pragma unroll
  for (int m = 16; m >= 1; m >>= 1) {
    s  += __shfl_xor(s, m, 32);
    s2 += __shfl_xor(s2, m, 32);
  }
  const float mu  = s * (1.0f / (float)HID);
  const float var = fmaxf(s2 * (1.0f / (float)HID) - mu * mu, 0.0f);
  const float inv = rsqrtf(var + LN_EPS);

  const float4 wa = reinterpret_cast<const float4*>(gw)[lane];
  const float4 wb = reinterpret_cast<const float4*>(gw)[lane + 32];
  const float4 ga = reinterpret_cast<const float4*>(gb)[lane];
  const float4 gb4 = reinterpret_cast<const float4*>(gb)[lane + 32];

  float w[8] = {wa.x, wa.y, wa.z, wa.w, wb.x, wb.y, wb.z, wb.w};
  float g[8] = {ga.x, ga.y, ga.z, ga.w, gb4.x, gb4.y, gb4.z, gb4.w};

  float y[8];
#pragma unroll
  for (int i = 0; i < 8; ++i) {
    float t = (v[i] - mu) * inv * w[i] + g[i];
    y[i] = (t >= 0.f) ? t : LEAKY * t;
  }
  float4 oa = {y[0], y[1], y[2], y[3]};
  float4 ob = {y[4], y[5], y[6], y[7]};
  reinterpret_cast<float4*>(out + (size_t)node * HID)[lane]      = oa;
  reinterpret_cast<float4*>(out + (size_t)node * HID)[lane + 32] = ob;
}

// --------------------------- pooling ---------------------------------------
__global__ void k_zero(float* __restrict__ out, float* __restrict__ cnt, int outN, int G) {
  int i = blockIdx.x * 256 + threadIdx.x;
  if (i < outN) out[i] = 0.0f;
  if (i < G)    cnt[i] = 0.0f;
}

__global__ __launch_bounds__(256)
void k_pool_scatter(const float* __restrict__ h, const int* __restrict__ batch,
                    float* __restrict__ sums, float* __restrict__ cnt, int N) {
  const int node = blockIdx.x * 8 + (threadIdx.x >> 5);
  if (node >= N) return;
  const int lane = threadIdx.x & 31;
  const int g = batch[node];
  const float4 xa = reinterpret_cast<const float4*>(h + (size_t)node * HID)[lane];
  const float4 xb = reinterpret_cast<const float4*>(h + (size_t)node * HID)[lane + 32];
  float* p = sums + (size_t)g * HID;
  atomAdd(p + (lane << 2) + 0,   xa.x);
  atomAdd(p + (lane << 2) + 1,   xa.y);
  atomAdd(p + (lane << 2) + 2,   xa.z);
  atomAdd(p + (lane << 2) + 3,   xa.w);
  atomAdd(p + 128 + (lane << 2) + 0, xb.x);
  atomAdd(p + 128 + (lane << 2) + 1, xb.y);
  atomAdd(p + 128 + (lane << 2) + 2, xb.z);
  atomAdd(p + 128 + (lane << 2) + 3, xb.w);
  if (lane == 0) atomAdd(&cnt[g], 1.0f);
}

__global__ void k_pool_div(float* __restrict__ out, const float* __restrict__ cnt, int G) {
  int i = blockIdx.x * 256 + threadIdx.x;
  if (i < G * HID) out[i] /= fmaxf(cnt[i / HID], 1.0f);
}

// --------------------------- launcher --------------------------------------
extern "C" void kernel_launch(void* const* d_in, const int* in_sizes, int n_in,
                              void* d_out, int out_size, void* d_ws, size_t ws_size,
                              hipStream_t stream) {
  const float* x    = (const float*)d_in[0];
  const int*   ei   = (const int*)d_in[1];      // [2][E] int32
  const int*   batch= (const int*)d_in[2];
  const float* W1   = (const float*)d_in[3];
  const float* b1   = (const float*)d_in[4];
  const float* ln1w = (const float*)d_in[5];
  const float* ln1b = (const float*)d_in[6];
  const float* W2   = (const float*)d_in[7];
  const float* b2   = (const float*)d_in[8];
  const float* ln2w = (const float*)d_in[9];
  const float* ln2b = (const float*)d_in[10];

  const int N = in_sizes[0] / HID;   // 50000 (divisible by 16)
  const int E = in_sizes[1] / 2;     // 800000
  const int G = out_size / HID;      // 512
  const int* src = ei;
  const int* dst = ei + E;

  float* ws   = (float*)d_ws;
  float* dinv = ws;                                   // N floats
  float* cnt  = ws + N;                               // G floats
  const size_t o = (size_t)((N + G + 3) & ~3);        // 16B align
  float* bufA = ws + o;                               // hs   [N*HID]
  float* bufB = bufA + (size_t)N * HID;               // acc  [N*HID]
  float* bufC = bufB + (size_t)N * HID;               // h1   [N*HID]
  float* outp = (float*)d_out;

  const int nodeBlk  = (N + 255) / 256;
  const int edgeBlk  = (E + 255) / 256;
  const int mtiles   = N / 16;
  const int scatBlk  = (E * 64 + 255) / 256;          // 1 thread per (edge, 4ch)
  const int wnodeBlk = (N + 7) / 8;                   // wave per node
  const int outBlk   = (out_size + 255) / 256;

  // degree -> dinv (shared by both layers)
  k_init_deg   <<<nodeBlk, 256, 0, stream>>>(dinv, N);
  k_deg_scatter<<<edgeBlk, 256, 0, stream>>>(dst, dinv, E);
  k_deg_to_dinv<<<nodeBlk, 256, 0, stream>>>(dinv, N);

  // layer 1
  gemm_rowscale <<<mtiles,  128, 0, stream>>>(x, W1, dinv, bufA, bufB);
  k_edge_scatter<<<scatBlk, 256, 0, stream>>>(src, dst, bufA, bufB, E);
  k_ln_act      <<<wnodeBlk,256, 0, stream>>>(bufB, dinv, b1, ln1w, ln1b, bufC, N);

  // layer 2 (input bufC; h2 -> bufA, safe since hs2 dead after scatter)
  gemm_rowscale <<<mtiles,  128, 0, stream>>>(bufC, W2, dinv, bufA, bufB);
  k_edge_scatter<<<scatBlk, 256, 0, stream>>>(src, dst, bufA, bufB, E);
  k_ln_act      <<<wnodeBlk,256, 0, stream>>>(bufB, dinv, b2, ln2w, ln2b, bufA, N);

  // global mean pool
  k_zero        <<<outBlk,  256, 0, stream>>>(outp, cnt, out_size, G);
  k_pool_scatter<<<wnodeBlk,256, 0, stream>>>(bufA, batch, outp, cnt, N);
  k_pool_div    <<<outBlk,  256, 0, stream>>>(outp, cnt, G);
}